// PNATower_2869038154151
// MI455X (gfx1250) — compile-verified
//
#include <hip/hip_runtime.h>
#include <hip/hip_bf16.h>

typedef float v2f __attribute__((ext_vector_type(2)));
typedef float v8f __attribute__((ext_vector_type(8)));

#define NB    8
#define NN    256
#define DIN   64
#define ROWS  (NB * NN)      // 2048
#define MCAT  832            // 13 * DIN
#define AVGDL 4.86f
#define EPSV  1e-5f
#define NEGB  1e9f

#if defined(__has_builtin)
#if __has_builtin(__builtin_amdgcn_global_load_async_to_lds_b128)
#define HAVE_ASYNC_LDS 1
#endif
#endif

#ifdef HAVE_ASYNC_LDS
typedef int v4i_gcc __attribute__((vector_size(16)));
typedef __attribute__((address_space(1))) v4i_gcc* gvp_t;  // global int4*
typedef __attribute__((address_space(3))) v4i_gcc* lvp_t;  // LDS int4*
#endif

__device__ __forceinline__ v8f wmma_f32(v2f a, v2f b, v8f c) {
  // V_WMMA_F32_16X16X4_F32: D = A(16x4) * B(4x16) + C(16x16), f32 throughout
  return __builtin_amdgcn_wmma_f32_16x16x4_f32(false, a, false, b, (short)0, c,
                                               false, false);
}

// ---------------------------------------------------------------------------
// Kernel 1: A = x @ W_pre[:64] + b_pre ; Q = x @ W_pre[64:]
// grid = 128 (16 rows each), block = 128 (4 waves; wave w owns col tile w)
// ---------------------------------------------------------------------------
__global__ void __launch_bounds__(128)
pna_pre(const float* __restrict__ x, const float* __restrict__ Wpre,
        const float* __restrict__ bpre, float* __restrict__ A,
        float* __restrict__ Q) {
  const int row0  = blockIdx.x * 16;
  const int lane  = threadIdx.x & 31;
  const int wave  = threadIdx.x >> 5;
  const int col   = (lane & 15) + wave * 16;
  const int arow  = row0 + (lane & 15);
  const int khalf = (lane >> 4) * 2;

  v8f accA = {};
  v8f accQ = {};
  for (int kk = 0; kk < DIN; kk += 4) {
    const int kb = kk + khalf;
    v2f a = *(const v2f*)(x + arow * DIN + kb);
    v2f bt, bb;
    bt.x = Wpre[kb * DIN + col];
    bt.y = Wpre[(kb + 1) * DIN + col];
    bb.x = Wpre[(DIN + kb) * DIN + col];
    bb.y = Wpre[(DIN + kb + 1) * DIN + col];
    accA = wmma_f32(a, bt, accA);
    accQ = wmma_f32(a, bb, accQ);
  }
  const float bp = bpre[col];
#pragma unroll
  for (int v = 0; v < 8; ++v) {
    const int gi = row0 + v + 8 * (lane >> 4);
    A[gi * DIN + col] = accA[v] + bp;  // fold b_pre into A
    Q[gi * DIN + col] = accQ[v];
  }
}

// ---------------------------------------------------------------------------
// Kernel 2: per 16-node tile: S1 = adj@Q, S2 = adj@Q^2 (WMMA), masked max/min
// of Q, degree scalers; writes m_cat (2048 x 832).
// adjacency tile staged to LDS via GLOBAL_LOAD_ASYNC_TO_LDS_B128 (ASYNCcnt).
// ---------------------------------------------------------------------------
__global__ void __launch_bounds__(128)
pna_agg(const float* __restrict__ x, const float* __restrict__ adj,
        const float* __restrict__ A, const float* __restrict__ Q,
        float* __restrict__ mcat) {
  __shared__ __align__(16) float adj_s[16 * NN];  // 16 KB contiguous rows
  __shared__ float A_s[16 * DIN];                 // 4 KB
  __shared__ float deg_s[16];

  const int row0 = blockIdx.x * 16;
  const int b    = row0 >> 8;
  const int i0   = row0 & 255;
  const int t    = threadIdx.x;

  const float* adjbase = adj + (size_t)b * NN * NN + (size_t)i0 * NN;
#ifdef HAVE_ASYNC_LDS
  {
#pragma unroll
    for (int k = 0; k < 8; ++k) {
      __builtin_amdgcn_global_load_async_to_lds_b128(
          (gvp_t)(adjbase + t * 4 + k * 512),
          (lvp_t)(adj_s + t * 4 + k * 512), 0, 0);
    }
  }
#else
  for (int k = t; k < (16 * NN) / 4; k += 128)
    ((float4*)adj_s)[k] = ((const float4*)adjbase)[k];
#endif
  for (int k = t; k < 16 * DIN; k += 128)
    A_s[k] = A[(row0 + (k >> 6)) * DIN + (k & 63)];
#ifdef HAVE_ASYNC_LDS
#if __has_builtin(__builtin_amdgcn_s_wait_asynccnt)
  __builtin_amdgcn_s_wait_asynccnt(0);
#else
  asm volatile("s_wait_asynccnt 0x0" ::: "memory");
#endif
#endif
  __syncthreads();
  if (t < 16) {
    float s = 0.f;
    for (int j = 0; j < NN; ++j) s += adj_s[t * NN + j];
    deg_s[t] = s;
  }
  __syncthreads();

  const int lane  = t & 31;
  const int wave  = t >> 5;
  const int col   = (lane & 15) + wave * 16;
  const int khalf = (lane >> 4) * 2;
  const float* Qb = Q + (size_t)b * NN * DIN;

  v8f acc1 = {};
  v8f acc2 = {};
  for (int kk = 0; kk < NN; kk += 4) {
    const int kb = kk + khalf;
    v2f a = *(const v2f*)(adj_s + (lane & 15) * NN + kb);
    v2f bq, bq2;
    bq.x  = Qb[kb * DIN + col];
    bq.y  = Qb[(kb + 1) * DIN + col];
    bq2.x = bq.x * bq.x;
    bq2.y = bq.y * bq.y;
    acc1 = wmma_f32(a, bq, acc1);   // adj @ Q
    acc2 = wmma_f32(a, bq2, acc2);  // adj @ Q^2
  }

  // masked max/min of Q over neighbors (rows rbase..rbase+7, fixed col)
  float mxq[8], mnq[8];
#pragma unroll
  for (int v = 0; v < 8; ++v) { mxq[v] = -3.0e38f; mnq[v] = 3.0e38f; }
  const int rbase = 8 * (lane >> 4);
  for (int j = 0; j < NN; ++j) {
    const float q = Qb[j * DIN + col];
#pragma unroll
    for (int v = 0; v < 8; ++v) {
      if (adj_s[(rbase + v) * NN + j] > 0.f) {
        mxq[v] = fmaxf(mxq[v], q);
        mnq[v] = fminf(mnq[v], q);
      }
    }
  }

#pragma unroll
  for (int v = 0; v < 8; ++v) {
    const int r  = rbase + v;
    const int gi = row0 + r;
    const float deg  = deg_s[r];
    const float dsafe = fmaxf(deg, 1.f);
    const float Av = A_s[r * DIN + col];
    const float S1 = acc1[v], S2 = acc2[v];
    const float s1   = deg * Av + S1;
    const float mean = s1 / dsafe;
    const float s2   = deg * Av * Av + 2.f * Av * S1 + S2;
    const float var  = fmaxf(s2 / dsafe - mean * mean, 0.f);
    const float sd   = sqrtf(var + EPSV);
    const bool  has  = deg > 0.f;
    const float mx   = has ? Av + mxq[v] : -NEGB;
    const float mn   = has ? Av + mnq[v] :  NEGB;
    const float logd = logf(deg + 1.f);
    const float ampf = logd / AVGDL;
    const float attf = AVGDL / fmaxf(logd, EPSV);
    float* row = mcat + (size_t)gi * MCAT;
    row[ 64 + col] = mean;        row[128 + col] = mx;
    row[192 + col] = mn;          row[256 + col] = sd;
    row[320 + col] = mean * ampf; row[384 + col] = mx * ampf;
    row[448 + col] = mn * ampf;   row[512 + col] = sd * ampf;
    row[576 + col] = mean * attf; row[640 + col] = mx * attf;
    row[704 + col] = mn * attf;   row[768 + col] = sd * attf;
  }
  // identity slot: m_cat[:, 0:64] = x
  for (int k = t; k < 16 * DIN; k += 128) {
    const int r = k >> 6, d = k & 63;
    mcat[(size_t)(row0 + r) * MCAT + d] = x[(row0 + r) * DIN + d];
  }
}

// ---------------------------------------------------------------------------
// Kernel 3: out = m_cat (2048x832) @ W_post (832x64) + b_post  (WMMA GEMM)
// ---------------------------------------------------------------------------
__global__ void __launch_bounds__(128)
pna_post(const float* __restrict__ mcat, const float* __restrict__ Wpost,
         const float* __restrict__ bpost, float* __restrict__ out) {
  const int row0  = blockIdx.x * 16;
  const int lane  = threadIdx.x & 31;
  const int wave  = threadIdx.x >> 5;
  const int col   = (lane & 15) + wave * 16;
  const int khalf = (lane >> 4) * 2;
  const float* arow = mcat + (size_t)(row0 + (lane & 15)) * MCAT;

  v8f acc = {};
  for (int kk = 0; kk < MCAT; kk += 4) {
    const int kb = kk + khalf;
    v2f a = *(const v2f*)(arow + kb);
    v2f bw;
    bw.x = Wpost[kb * 64 + col];
    bw.y = Wpost[(kb + 1) * 64 + col];
    acc = wmma_f32(a, bw, acc);
  }
  const float bp = bpost[col];
#pragma unroll
  for (int v = 0; v < 8; ++v) {
    const int gi = row0 + v + 8 * (lane >> 4);
    out[(size_t)gi * 64 + col] = acc[v] + bp;
  }
}

extern "C" void kernel_launch(void* const* d_in, const int* in_sizes, int n_in,
                              void* d_out, int out_size, void* d_ws, size_t ws_size,
                              hipStream_t stream) {
  (void)in_sizes; (void)n_in; (void)out_size; (void)ws_size;
  const float* x     = (const float*)d_in[0];
  const float* adj   = (const float*)d_in[1];
  const float* Wpre  = (const float*)d_in[2];
  const float* bpre  = (const float*)d_in[3];
  const float* Wpost = (const float*)d_in[4];
  const float* bpost = (const float*)d_in[5];
  float* out = (float*)d_out;

  float* A    = (float*)d_ws;            // 2048*64 f32
  float* Q    = A + ROWS * DIN;          // 2048*64 f32
  float* mcat = Q + ROWS * DIN;          // 2048*832 f32  (total ~7.5 MB)

  dim3 blk(128);
  dim3 grd(ROWS / 16);
  pna_pre <<<grd, blk, 0, stream>>>(x, Wpre, bpre, A, Q);
  pna_agg <<<grd, blk, 0, stream>>>(x, adj, A, Q, mcat);
  pna_post<<<grd, blk, 0, stream>>>(mcat, Wpost, bpost, out);
}